// ConvolutionBlock_21517786153335
// MI455X (gfx1250) — compile-verified
//
#include <hip/hip_runtime.h>

typedef __attribute__((ext_vector_type(2))) float v2f;
typedef __attribute__((ext_vector_type(8))) float v8f;

#define NPTS 400000
#define INC  32
#define OUTC 32
#define KOFF 27
#define TILES (NPTS / 16)   // 25000, exact

// ---------------------------------------------------------------------------
// Kernel 0: zero the accumulation target (d_out is poisoned by the harness).
// ---------------------------------------------------------------------------
__global__ void zero_out_kernel(float* __restrict__ out, int n4) {
    int i = blockIdx.x * blockDim.x + threadIdx.x;
    if (i < n4) {
        ((float4*)out)[i] = make_float4(0.f, 0.f, 0.f, 0.f);
    }
}

// ---------------------------------------------------------------------------
// Kernel 1: gather -> WMMA f32 GEMM -> scatter atomic-add.
// blockIdx.y = offset k (27), blockIdx.x = tile chunk; 256 thr = 8 waves,
// one 16-point tile per wave per iteration.
// ---------------------------------------------------------------------------
__global__ __launch_bounds__(256) void spconv_wmma_kernel(
    const float* __restrict__ feats,     // [NPTS, 32]
    const float* __restrict__ weights,   // [27, 32, 32] (ic-major, oc minor)
    const int*   __restrict__ in_idx,    // [27, NPTS]
    const int*   __restrict__ out_idx,   // [27, NPTS]
    float*       __restrict__ out)       // [NPTS, 32] accumulator
{
    const int k    = blockIdx.y;
    const int lane = threadIdx.x & 31;
    const int wave = threadIdx.x >> 5;
    const int lo   = lane & 15;   // column / row-within-half
    const int hi   = lane >> 4;   // K-pair select (A), K-row select (B)

    // Build B fragments for W_k once per wave; W_k is 4 KB and L2-resident.
    // B is KxN = 32x32 split into two 16-col halves; 8 K-steps of 4.
    // Guessed layout mirrors the A 16x4 table: VGPR0 holds K={4s, 4s+2},
    // VGPR1 holds K={4s+1, 4s+3}; row striped across lanes (N = lane%16).
    const float* Wk = weights + k * (INC * OUTC);
    v2f b0[8], b1[8];
#pragma unroll
    for (int s = 0; s < 8; ++s) {
        const int r0 = 4 * s + 2 * hi;
        b0[s].x = Wk[(r0    ) * OUTC + lo];
        b0[s].y = Wk[(r0 + 1) * OUTC + lo];
        b1[s].x = Wk[(r0    ) * OUTC + 16 + lo];
        b1[s].y = Wk[(r0 + 1) * OUTC + 16 + lo];
    }

    const int* iin_k  = in_idx  + (long)k * NPTS;
    const int* iout_k = out_idx + (long)k * NPTS;

    const int wstride = gridDim.x * (256 / 32);
    for (int t = blockIdx.x * 8 + wave; t < TILES; t += wstride) {
        const int base = t * 16;

        // Irregular gather: lane pair (L, L+16) covers row lo, 64B apiece.
        const int iin = iin_k[base + lo];
        const float* arow = feats + (long)iin * INC;

        v8f c0 = {};
        v8f c1 = {};
#pragma unroll
        for (int s = 0; s < 8; ++s) {
            // A frag: lane holds feats[row lo][4s + 2*hi .. +1] (8B load).
            v2f a = *(const v2f*)(arow + 4 * s + 2 * hi);
            c0 = __builtin_amdgcn_wmma_f32_16x16x4_f32(
                     false, a, false, b0[s], (short)0, c0, false, false);
            c1 = __builtin_amdgcn_wmma_f32_16x16x4_f32(
                     false, a, false, b1[s], (short)0, c1, false, false);
        }

        // Scatter-accumulate: D VGPR r, lane L -> M = r + 8*hi, N = lo.
#pragma unroll
        for (int r = 0; r < 8; ++r) {
            const int orow = iout_k[base + r + 8 * hi];
            float* dst = out + (long)orow * OUTC + lo;
            atomicAdd(dst,      c0[r]);
            atomicAdd(dst + 16, c1[r]);
        }
    }
}

// ---------------------------------------------------------------------------
// Kernel 2: in-place BatchNorm (eval) + ReLU.
// ---------------------------------------------------------------------------
__global__ void bn_relu_kernel(float* __restrict__ out,
                               const float* __restrict__ gamma,
                               const float* __restrict__ beta,
                               const float* __restrict__ mean,
                               const float* __restrict__ var,
                               int n)
{
    int i = blockIdx.x * blockDim.x + threadIdx.x;
    if (i < n) {
        const int c = i & (OUTC - 1);
        const float inv = rsqrtf(var[c] + 1e-5f);
        const float val = (out[i] - mean[c]) * (gamma[c] * inv) + beta[c];
        out[i] = fmaxf(val, 0.0f);
    }
}

// ---------------------------------------------------------------------------
extern "C" void kernel_launch(void* const* d_in, const int* in_sizes, int n_in,
                              void* d_out, int out_size, void* d_ws, size_t ws_size,
                              hipStream_t stream) {
    const float* feats   = (const float*)d_in[0];
    const float* weights = (const float*)d_in[1];
    const float* gamma   = (const float*)d_in[2];
    const float* beta    = (const float*)d_in[3];
    const float* mean    = (const float*)d_in[4];
    const float* var     = (const float*)d_in[5];
    const int*   in_idx  = (const int*)d_in[6];
    const int*   out_idx = (const int*)d_in[7];
    float* out = (float*)d_out;

    const int n  = NPTS * OUTC;        // 12.8M floats
    const int n4 = n / 4;

    zero_out_kernel<<<(n4 + 255) / 256, 256, 0, stream>>>(out, n4);

    dim3 grid(250, KOFF);              // 250 blocks * 8 waves = 2000 waves/offset
    spconv_wmma_kernel<<<grid, 256, 0, stream>>>(feats, weights, in_idx, out_idx, out);

    bn_relu_kernel<<<(n + 255) / 256, 256, 0, stream>>>(out, gamma, beta, mean, var, n);
}